// GCN_JK_Concat_43731357008167
// MI455X (gfx1250) — compile-verified
//
#include <hip/hip_runtime.h>
#include <hip/hip_bf16.h>

// ---------------------------------------------------------------------------
// GCN (3x GCNConv + JK-concat linear) for MI455X / gfx1250, wave32.
// - Dense GEMMs on V_WMMA_F32_16X16X4_F32 (full-f32 matrix pipe).
// - Weights pre-packed so each WMMA B-fragment is ONE ds_load_b64 (K-pairs
//   interleaved by column), eliminating the v_mov repack chains.
// - W tiles staged into LDS with global_load_async_to_lds_b128 (ASYNCcnt
//   path, s_wait_asynccnt) instead of VGPR round-trips.
// - Sparse aggregation via f32 global atomics; working set (~103 MB) fits
//   the 192 MB L2 so gathers/atomics stay on-chip.
// ---------------------------------------------------------------------------

typedef __attribute__((ext_vector_type(2))) float v2f;
typedef __attribute__((ext_vector_type(8))) float v8f;

#ifndef HFEAT
#define HFEAT 64
#endif

// Raw 32-bit LDS byte offset of a __shared__ object (addrspace(3) ptrtoint).
__device__ __forceinline__ unsigned lds_offset(const void* p) {
  return (unsigned)(unsigned long long)(__attribute__((address_space(3))) const void*)p;
}

// Async 16-byte copy: global -> LDS, tracked by ASYNCcnt.
__device__ __forceinline__ void async_copy16(unsigned lds_byte_off, const void* gptr) {
  asm volatile("global_load_async_to_lds_b128 %0, %1, off"
               :
               : "v"(lds_byte_off), "v"(gptr)
               : "memory");
}

__device__ __forceinline__ void wait_async() {
  asm volatile("s_wait_asynccnt 0" ::: "memory");
}

// ---- degree / normalization -----------------------------------------------

__global__ void k_fill_ones(float* __restrict__ p, int n) {
  int i = blockIdx.x * blockDim.x + threadIdx.x;
  if (i < n) p[i] = 1.0f;  // self-loop contributes 1 to every node's degree
}

__global__ void k_deg_edges(const int* __restrict__ dst, float* __restrict__ deg, int E) {
  int i = blockIdx.x * blockDim.x + threadIdx.x;
  if (i < E) atomicAdd(&deg[dst[i]], 1.0f);
}

__global__ void k_finish_dis(float* __restrict__ p, int n) {
  int i = blockIdx.x * blockDim.x + threadIdx.x;
  if (i < n) p[i] = rsqrtf(p[i]);  // deg >= 1 always (self loops)
}

// ---- weight packing --------------------------------------------------------
// Wp[(k>>1)*128 + n*2 + (k&1)] = W[k*64 + n]  -> B fragment {W[k][n],W[k+1][n]}
// becomes one aligned 8-byte load.

__global__ void k_pack_w(const float* __restrict__ W, float* __restrict__ Wp, int total) {
  int i = blockIdx.x * blockDim.x + threadIdx.x;
  if (i < total) {
    int k = i >> 6, n = i & 63;
    Wp[(k >> 1) * 128 + n * 2 + (k & 1)] = W[i];
  }
}

// ---- aggregation -----------------------------------------------------------

// h[v] = xw[v] * dis[v]^2   (self-loop term, also zero-initializes h)
__global__ void k_init_self(const float* __restrict__ xw, const float* __restrict__ dis,
                            float* __restrict__ h, int n64) {
  int i = blockIdx.x * blockDim.x + threadIdx.x;
  if (i < n64) {
    float d = dis[i >> 6];
    h[i] = xw[i] * d * d;
  }
}

// h[dst] += xw[src] * dis[src]*dis[dst]; 16 threads/edge, float4 per thread.
__global__ void k_scatter_edges(const float* __restrict__ xw, const int* __restrict__ src,
                                const int* __restrict__ dst, const float* __restrict__ dis,
                                float* __restrict__ h, long long E) {
  long long t = (long long)blockIdx.x * blockDim.x + threadIdx.x;
  long long e = t >> 4;
  if (e >= E) return;
  int f = ((int)t & 15) * 4;
  int s = src[e];
  int d = dst[e];
  float nrm = dis[s] * dis[d];
  const float4 v = *(const float4*)(xw + (long long)s * HFEAT + f);
  float* o = h + (long long)d * HFEAT + f;
  atomicAdd(o + 0, v.x * nrm);
  atomicAdd(o + 1, v.y * nrm);
  atomicAdd(o + 2, v.z * nrm);
  atomicAdd(o + 3, v.w * nrm);
}

__global__ void k_relu_bias(float* __restrict__ h, const float* __restrict__ b, int n64) {
  int i = blockIdx.x * blockDim.x + threadIdx.x;
  if (i < n64) h[i] = fmaxf(h[i] + b[i & 63], 0.0f);
}

// ---- dense GEMM on the f32 matrix pipe ------------------------------------
// out[N x 64] = A[N x K] @ W[K x 64]. One 256-thread block covers 128 rows
// (8 waves x 16-row strip); packed W async-staged into LDS.
//
// V_WMMA_F32_16X16X4_F32 fragment layouts (wave32):
//   A (16x4):  lane half h, col c: a[r] = A[M=c][K = k0 + 2h + r],  r=0,1
//   B (4x16):  b[r] = B[K = k0 + 2h + r][N=c]   (one ds_load_b64 from Wp)
//   C/D:       acc[r] = D[M = 8h + r][N=c], r=0..7

template <int K>
__global__ __launch_bounds__(256) void k_gemm_wmma(const float* __restrict__ A,
                                                   const float* __restrict__ Wp,
                                                   float* __restrict__ out, int nrows) {
  __shared__ float wlds[K * 64];
  {
    const unsigned base = lds_offset(wlds);
    const char* g = (const char*)Wp;
    for (int c = threadIdx.x; c < K * 16; c += 256)  // 16B chunks
      async_copy16(base + (unsigned)c * 16u, g + (size_t)c * 16);
    wait_async();
  }
  __syncthreads();

  const int lane = threadIdx.x & 31;
  const int wave = threadIdx.x >> 5;
  const int half = lane >> 4;
  const int col = lane & 15;
  const long long rowbase = (long long)blockIdx.x * 128 + wave * 16;
  if (rowbase >= nrows) return;  // wave-uniform exit (after the only barrier)

  long long arow = rowbase + col;
  if (arow > (long long)nrows - 1) arow = nrows - 1;  // clamp: EXEC must stay all-1s
  const float* aptr = A + arow * K;

  v8f acc[4] = {};
  for (int k0 = 0; k0 < K; k0 += 4) {
    const int kk = k0 + 2 * half;  // even
    v2f a = *(const v2f*)(aptr + kk);
#pragma unroll
    for (int nt = 0; nt < 4; ++nt) {
      v2f b = *(const v2f*)&wlds[(kk >> 1) * 128 + (nt * 16 + col) * 2];
      acc[nt] = __builtin_amdgcn_wmma_f32_16x16x4_f32(false, a, false, b, (short)0, acc[nt],
                                                      false, false);
    }
  }

#pragma unroll
  for (int nt = 0; nt < 4; ++nt) {
#pragma unroll
    for (int r = 0; r < 8; ++r) {
      long long row = rowbase + half * 8 + r;
      if (row < nrows) out[row * 64 + nt * 16 + col] = acc[nt][r];
    }
  }
}

// ---- fused JK GEMM: out = [x | h1 | h2 | h3] @ Wl + bl --------------------
// Concat never materialized: 4 K-segments (128,64,64,64) against row-slices
// of packed Wl, each slice async-staged into <=32 KB LDS, shared accumulators.

__global__ __launch_bounds__(256) void k_final_gemm(const float* __restrict__ x,
                                                    const float* __restrict__ h1,
                                                    const float* __restrict__ h2,
                                                    const float* __restrict__ h3,
                                                    const float* __restrict__ Wlp,
                                                    const float* __restrict__ bl,
                                                    float* __restrict__ out, int nrows) {
  __shared__ float wlds[128 * 64];

  const int lane = threadIdx.x & 31;
  const int wave = threadIdx.x >> 5;
  const int half = lane >> 4;
  const int col = lane & 15;
  const long long rowbase = (long long)blockIdx.x * 128 + wave * 16;
  const bool active = rowbase < nrows;  // keep every wave alive across barriers

  long long arow = rowbase + col;
  if (arow > (long long)nrows - 1) arow = nrows - 1;
  if (arow < 0) arow = 0;

  const float* srcs[4] = {x, h1, h2, h3};
  const int ks[4] = {128, 64, 64, 64};
  const unsigned base = lds_offset(wlds);
  v8f acc[4] = {};

  int woff = 0;
  for (int s = 0; s < 4; ++s) {
    __syncthreads();  // previous segment's LDS fully consumed
    const int kseg = ks[s];
    // packed rows for this segment start at (woff/2)*128 floats
    const char* g = (const char*)(Wlp + (size_t)(woff >> 1) * 128);
    for (int c = threadIdx.x; c < kseg * 16; c += 256)  // 16B chunks
      async_copy16(base + (unsigned)c * 16u, g + (size_t)c * 16);
    wait_async();
    __syncthreads();

    if (active) {  // wave-uniform branch: EXEC all-1s inside for WMMA
      const float* aptr = srcs[s] + arow * ks[s];
      for (int k0 = 0; k0 < kseg; k0 += 4) {
        const int kk = k0 + 2 * half;  // even
        v2f a = *(const v2f*)(aptr + kk);
#pragma unroll
        for (int nt = 0; nt < 4; ++nt) {
          v2f b = *(const v2f*)&wlds[(kk >> 1) * 128 + (nt * 16 + col) * 2];
          acc[nt] = __builtin_amdgcn_wmma_f32_16x16x4_f32(false, a, false, b, (short)0, acc[nt],
                                                          false, false);
        }
      }
    }
    woff += kseg;
  }

  if (active) {
#pragma unroll
    for (int nt = 0; nt < 4; ++nt) {
      float bias = bl[nt * 16 + col];
#pragma unroll
      for (int r = 0; r < 8; ++r) {
        long long row = rowbase + half * 8 + r;
        if (row < nrows) out[row * 64 + nt * 16 + col] = acc[nt][r] + bias;
      }
    }
  }
}

// ---------------------------------------------------------------------------

extern "C" void kernel_launch(void* const* d_in, const int* in_sizes, int n_in,
                              void* d_out, int out_size, void* d_ws, size_t ws_size,
                              hipStream_t stream) {
  const float* x = (const float*)d_in[0];
  const int* ei = (const int*)d_in[1];
  const float* W0 = (const float*)d_in[2];
  const float* b0 = (const float*)d_in[3];
  const float* W1 = (const float*)d_in[4];
  const float* b1 = (const float*)d_in[5];
  const float* W2 = (const float*)d_in[6];
  const float* b2 = (const float*)d_in[7];
  const float* Wl = (const float*)d_in[8];
  const float* bl = (const float*)d_in[9];
  float* out = (float*)d_out;

  const int N = in_sizes[0] / 128;
  const long long E = in_sizes[1] / 2;
  const int* src = ei;      // edge_index[0]
  const int* dst = ei + E;  // edge_index[1]

  // workspace (f32): dis[N] | xw | h1 | h2 | h3 | W0p | W1p | W2p | Wlp
  float* dis = (float*)d_ws;
  float* xw = dis + (((size_t)N + 255) & ~(size_t)255);
  float* h1 = xw + (size_t)N * 64;
  float* h2 = h1 + (size_t)N * 64;
  float* h3 = h2 + (size_t)N * 64;
  float* W0p = h3 + (size_t)N * 64;
  float* W1p = W0p + 128 * 64;
  float* W2p = W1p + 64 * 64;
  float* Wlp = W2p + 64 * 64;

  const int n64 = N * 64;
  auto cdiv = [](long long a, long long b) { return (unsigned)((a + b - 1) / b); };
  const dim3 blk(256);
  const unsigned gN = cdiv(N, 256), g64 = cdiv(n64, 256);
  const unsigned gE = cdiv(E, 256), gS = cdiv(E * 16, 256);
  const unsigned gG = cdiv(N, 128);

  // pack weights into K-pair-interleaved layout (one-time, tiny)
  k_pack_w<<<cdiv(128 * 64, 256), blk, 0, stream>>>(W0, W0p, 128 * 64);
  k_pack_w<<<cdiv(64 * 64, 256), blk, 0, stream>>>(W1, W1p, 64 * 64);
  k_pack_w<<<cdiv(64 * 64, 256), blk, 0, stream>>>(W2, W2p, 64 * 64);
  k_pack_w<<<cdiv(320 * 64, 256), blk, 0, stream>>>(Wl, Wlp, 320 * 64);

  // D^{-1/2} with self loops
  k_fill_ones<<<gN, blk, 0, stream>>>(dis, N);
  k_deg_edges<<<gE, blk, 0, stream>>>(dst, dis, (int)E);
  k_finish_dis<<<gN, blk, 0, stream>>>(dis, N);

  // layer 0: x@W0 -> aggregate -> relu(+b0)
  k_gemm_wmma<128><<<gG, blk, 0, stream>>>(x, W0p, xw, N);
  k_init_self<<<g64, blk, 0, stream>>>(xw, dis, h1, n64);
  k_scatter_edges<<<gS, blk, 0, stream>>>(xw, src, dst, dis, h1, E);
  k_relu_bias<<<g64, blk, 0, stream>>>(h1, b0, n64);

  // layer 1
  k_gemm_wmma<64><<<gG, blk, 0, stream>>>(h1, W1p, xw, N);
  k_init_self<<<g64, blk, 0, stream>>>(xw, dis, h2, n64);
  k_scatter_edges<<<gS, blk, 0, stream>>>(xw, src, dst, dis, h2, E);
  k_relu_bias<<<g64, blk, 0, stream>>>(h2, b1, n64);

  // layer 2
  k_gemm_wmma<64><<<gG, blk, 0, stream>>>(h2, W2p, xw, N);
  k_init_self<<<g64, blk, 0, stream>>>(xw, dis, h3, n64);
  k_scatter_edges<<<gS, blk, 0, stream>>>(xw, src, dst, dis, h3, E);
  k_relu_bias<<<g64, blk, 0, stream>>>(h3, b2, n64);

  // JK concat + final linear, fused
  k_final_gemm<<<gG, blk, 0, stream>>>(x, h1, h2, h3, Wlp, bl, out, N);
}